// LinkedCrossEntropy_35785667510369
// MI455X (gfx1250) — compile-verified
//
#include <hip/hip_runtime.h>
#include <cstdint>
#include <cstddef>

// ---------------------------------------------------------------------------
// LinkedCrossEntropy for MI455X (gfx1250, wave32).
//
// Memory-bound: 262 MB of y_pred read once -> ~11.3 us at 23.3 TB/s HBM.
// One wave per row; row (1000 f32 = 250 float4, 16B-aligned base) is staged
// via the CDNA5 async global->LDS engine (GLOBAL_LOAD_ASYNC_TO_LDS_B128 +
// s_wait_asynccnt), then reduced in registers (max/argmax pass + sum-exp
// pass) with wave32 shfl_xor trees. Deterministic two-stage block-sum
// reduction (no float atomics).
// ---------------------------------------------------------------------------

#if defined(__gfx1250__) && __has_builtin(__builtin_amdgcn_global_load_async_to_lds_b128)
#define USE_ASYNC_LDS 1
#else
#define USE_ASYNC_LDS 0
#endif

namespace {

constexpr int   kC             = 1000;        // classes (reference fixes C=1000)
constexpr int   kNV4           = kC / 4;      // 250 float4 per row
constexpr int   kTiles         = 8;           // ceil(250/32)
constexpr int   kWavesPerBlock = 8;
constexpr int   kBlockThreads  = kWavesPerBlock * 32;
constexpr float kAlphaLink     = 2.0f;

} // namespace

#if USE_ASYNC_LDS
// Builtin signature (leaked by clang diagnostic): param 1 is a non-const
// int4 pointer in addrspace(1); param 2 the LDS-side pointer in addrspace(3).
typedef int v4i_t __attribute__((ext_vector_type(4)));
typedef __attribute__((address_space(1))) v4i_t* gv4_ptr;
typedef __attribute__((address_space(3))) v4i_t* lv4_ptr;
#endif

__launch_bounds__(kBlockThreads)
__global__ void linkedce_row_kernel(const float* __restrict__ y_pred,
                                    const int* __restrict__ y_true,
                                    const float* __restrict__ weight,
                                    const unsigned char* __restrict__ link,  // bool [C,C]
                                    float* __restrict__ block_sums,
                                    int nrows) {
    const int wave = threadIdx.x >> 5;
    const int lane = threadIdx.x & 31;
    const int row  = blockIdx.x * kWavesPerBlock + wave;
    const float NEG_INF = -__builtin_inff();

    __shared__ float wave_loss[kWavesPerBlock];
#if USE_ASYNC_LDS
    // Per-wave staging buffer: 250 float4 padded to 256 (4 KB) -> 32 KB/block.
    __shared__ float4 abuf[kWavesPerBlock][256];
#endif

    if (row < nrows) {
        const float* xr = y_pred + (size_t)row * kC;
        float4 v[kTiles];

#if USE_ASYNC_LDS
        {
            float4* src = const_cast<float4*>((const float4*)xr);
#pragma unroll
            for (int k = 0; k < kTiles; ++k) {
                const int j4 = lane + (k << 5);
                if (j4 < kNV4) {
                    __builtin_amdgcn_global_load_async_to_lds_b128(
                        (gv4_ptr)(src + j4),
                        (lv4_ptr)(&abuf[wave][j4]),
                        /*offset=*/0, /*cpol=*/0);
                }
            }
        }
#if __has_builtin(__builtin_amdgcn_s_wait_asynccnt)
        __builtin_amdgcn_s_wait_asynccnt(0);
#else
        asm volatile("s_wait_asynccnt 0x0" ::: "memory");
#endif
        asm volatile("" ::: "memory");
#pragma unroll
        for (int k = 0; k < kTiles; ++k) {
            const int j4 = lane + (k << 5);
            v[k] = (j4 < kNV4) ? abuf[wave][j4]
                               : make_float4(NEG_INF, NEG_INF, NEG_INF, NEG_INF);
        }
#else
#pragma unroll
        for (int k = 0; k < kTiles; ++k) {
            const int j4 = lane + (k << 5);
            v[k] = (j4 < kNV4) ? ((const float4*)xr)[j4]
                               : make_float4(NEG_INF, NEG_INF, NEG_INF, NEG_INF);
        }
#endif

        // ---- pass 1: per-lane max + argmax (indices ascend -> '>' keeps first)
        float bm = NEG_INF;
        int   bi = 0;
#pragma unroll
        for (int k = 0; k < kTiles; ++k) {
            const int base = (lane + (k << 5)) << 2;
            if (v[k].x > bm) { bm = v[k].x; bi = base;     }
            if (v[k].y > bm) { bm = v[k].y; bi = base + 1; }
            if (v[k].z > bm) { bm = v[k].z; bi = base + 2; }
            if (v[k].w > bm) { bm = v[k].w; bi = base + 3; }
        }
        // wave32 reduction; tie-break to lower index = jnp.argmax semantics
#pragma unroll
        for (int m = 16; m >= 1; m >>= 1) {
            const float om = __shfl_xor(bm, m, 32);
            const int   oi = __shfl_xor(bi, m, 32);
            if (om > bm || (om == bm && oi < bi)) { bm = om; bi = oi; }
        }

        // ---- pass 2: sum of exp(x - rowmax); -inf pads contribute exactly 0
        float s = 0.0f;
#pragma unroll
        for (int k = 0; k < kTiles; ++k) {
            s += __expf(v[k].x - bm);
            s += __expf(v[k].y - bm);
            s += __expf(v[k].z - bm);
            s += __expf(v[k].w - bm);
        }
#pragma unroll
        for (int m = 16; m >= 1; m >>= 1) s += __shfl_xor(s, m, 32);

        if (lane == 0) {
            const int   t   = y_true[row];
            const float xt  = xr[t];
            const float nll = bm + __logf(s) - xt;   // -(xt - max - log(sumexp))
            float pen = 1.0f;
            if (bi != t) {
                if (link[(size_t)t * kC + bi]) pen = kAlphaLink;
            }
            wave_loss[wave] = pen * weight[t] * nll;
        }
    } else if (lane == 0) {
        wave_loss[wave] = 0.0f;
    }

    __syncthreads();
    if (threadIdx.x == 0) {
        float acc = 0.0f;
#pragma unroll
        for (int w = 0; w < kWavesPerBlock; ++w) acc += wave_loss[w];
        block_sums[blockIdx.x] = acc;
    }
}

__launch_bounds__(256)
__global__ void linkedce_final_reduce(const float* __restrict__ part,
                                      float* __restrict__ out,
                                      int n, float inv_b) {
    __shared__ float sdata[256];
    float s = 0.0f;
    for (int i = threadIdx.x; i < n; i += 256) s += part[i];
    sdata[threadIdx.x] = s;
    __syncthreads();
    for (int off = 128; off >= 32; off >>= 1) {
        if (threadIdx.x < off) sdata[threadIdx.x] += sdata[threadIdx.x + off];
        __syncthreads();
    }
    if (threadIdx.x < 32) {
        float v = sdata[threadIdx.x];
#pragma unroll
        for (int m = 16; m >= 1; m >>= 1) v += __shfl_xor(v, m, 32);
        if (threadIdx.x == 0) out[0] = v * inv_b;
    }
}

extern "C" void kernel_launch(void* const* d_in, const int* in_sizes, int n_in,
                              void* d_out, int out_size, void* d_ws, size_t ws_size,
                              hipStream_t stream) {
    (void)n_in; (void)out_size; (void)ws_size;
    const float*         y_pred = (const float*)d_in[0];
    const int*           y_true = (const int*)d_in[1];
    const float*         weight = (const float*)d_in[2];
    const unsigned char* link   = (const unsigned char*)d_in[3];  // jnp.bool_ = 1 byte
    float* out = (float*)d_out;

    const int B = in_sizes[1];                 // 65536 rows (y_true length)
    float* block_sums = (float*)d_ws;          // nblocks floats (32 KB) scratch

    const int nblocks = (B + kWavesPerBlock - 1) / kWavesPerBlock;
    linkedce_row_kernel<<<nblocks, kBlockThreads, 0, stream>>>(
        y_pred, y_true, weight, link, block_sums, B);
    linkedce_final_reduce<<<1, 256, 0, stream>>>(
        block_sums, out, nblocks, 1.0f / (float)B);
}